// DilateAttention_28226525069829
// MI455X (gfx1250) — compile-verified
//
#include <hip/hip_runtime.h>

#define HEADS 6
#define HD    32
#define HW    112
#define LPIX  (HW*HW)          // 12544
#define TS    16               // tile side (pixels)
#define CH    20               // container rows  = TS + 2*2 halo
#define CW    24               // container cols  = TS + 8 (16B-aligned halo)
#define TILE_F (HD*CH*CW)      // 15360 floats per tensor tile
#define SCALE 0.17677669529663687f   // 32^-0.5

typedef __attribute__((address_space(3))) float* lds_fp;
typedef __attribute__((ext_vector_type(4))) unsigned int u32x4;
typedef __attribute__((ext_vector_type(8))) int i32x8;
typedef __attribute__((ext_vector_type(4))) int i32x4;

// One TDM descriptor moves a whole [32][20][24] f32 halo tile (61.4 KB)
// global -> LDS, packed contiguously (X fastest, row stride = tile_dim0).
__device__ __forceinline__ void tdm_load_tile(const float* gsrc, unsigned lds_byte_off)
{
    unsigned long long ga = (unsigned long long)(uintptr_t)gsrc;

    u32x4 g0;
    g0[0] = 1u;                                               // count=1 (valid), user mode
    g0[1] = lds_byte_off;                                     // lds_addr
    g0[2] = (unsigned)(ga & 0xffffffffu);                     // global_addr[31:0]
    g0[3] = (unsigned)((ga >> 32) & 0x1ffffffu) | (2u << 30); // global_addr[56:32] | type=2

    i32x8 g1;
    g1[0] = (int)(2u << 16);           // wg_mask=0 (not in cluster), data_size=4B
    g1[1] = (int)(112u << 16);         // atomic_barrier=0 | tensor_dim0[15:0]=112
    g1[2] = (int)(112u << 16);         // tensor_dim0[31:16]=0 | tensor_dim1[15:0]=112
    g1[3] = (int)(24u << 16);          // tensor_dim1[31:16]=0 | tile_dim0=24
    g1[4] = (int)(20u | (32u << 16));  // tile_dim1=20 | tile_dim2=32
    g1[5] = 112;                       // tensor_dim0_stride = 112 (row)
    g1[6] = (int)(12544u << 16);       // stride0[47:32]=0 | tensor_dim1_stride[15:0]=12544
    g1[7] = 0;                         // tensor_dim1_stride[47:16]=0

    i32x4 g2;
    g2[0] = 32;                        // tensor_dim2 (d planes)
    g2[1] = 0;                         // tensor_dim3 (unused)
    g2[2] = 0;                         // tensor_dim2_stride (unused, 3D tile)
    g2[3] = 0;                         // stride2 hi | tile_dim3=0
    i32x4 g3 = {0, 0, 0, 0};           // dims 3/4 unused

    i32x8 gx = {0, 0, 0, 0, 0, 0, 0, 0};  // extra operand (clang-23 6-arg form)

    __builtin_amdgcn_tensor_load_to_lds(g0, g1, g2, g3, gx, 0);
}

__global__ __launch_bounds__(256)
void dilate_attn_kernel(const float* __restrict__ q,
                        const float* __restrict__ k,
                        const float* __restrict__ v,
                        float* __restrict__ out)
{
    __shared__ float smem[2 * TILE_F];   // [0..): K tile, [TILE_F..): V tile

    const int tid  = threadIdx.x;
    const int x0   = blockIdx.x * TS;
    const int y0   = blockIdx.y * TS;
    const int head = blockIdx.z % HEADS;
    const int b    = blockIdx.z / HEADS;

    const size_t chan0 = (size_t)(b * (HEADS * HD) + head * HD);
    const bool border = (x0 == 0) | (y0 == 0) | (x0 == HW - TS) | (y0 == HW - TS);

    // ---- issue Q register loads FIRST: they overlap with TDM/async staging ----
    const int tx = tid & (TS - 1);
    const int ty = tid >> 4;
    const int gx = x0 + tx;
    const int gy = y0 + ty;

    const float* qp = q + chan0 * LPIX + gy * HW + gx;
    float qr[HD];
    #pragma unroll
    for (int d = 0; d < HD; ++d) qr[d] = qp[(size_t)d * LPIX];

    if (!border) {
        // ---- interior tile: Tensor Data Mover stages K (wave 0) and V (wave 1) ----
        if (tid < 64) {
            const bool   isK = (tid < 32);
            const float* src = isK ? k : v;
            const float* gp  = src + chan0 * LPIX + (y0 - 2) * HW + (x0 - 4);
            lds_fp   lp      = (lds_fp)&smem[isK ? 0 : TILE_F];
            unsigned ldsoff  = (unsigned)(uintptr_t)lp;
            tdm_load_tile(gp, ldsoff);
        }
        __builtin_amdgcn_s_wait_tensorcnt((short)0);
        __syncthreads();
    } else {
        // ---- border tile: zero-init (zero padding), then per-lane async copies ----
        {
            const float4 z = make_float4(0.f, 0.f, 0.f, 0.f);
            #pragma unroll
            for (int it = 0; it < (2 * TILE_F / 4) / 256; ++it)
                *(float4*)&smem[(it * 256 + tid) * 4] = z;
        }
        __syncthreads();   // ds stores complete before async writes land

        #pragma unroll 1
        for (int t = 0; t < 2; ++t) {
            const float* src = (t ? v : k) + chan0 * LPIX;
            float*       dst = &smem[t * TILE_F];
            #pragma unroll 1
            for (int it = 0; it < 15; ++it) {
                int i   = it * 256 + tid;
                int d   = i / (CH * 6);
                int rem = i - d * (CH * 6);
                int r   = rem / 6;
                int s4  = rem - r * 6;
                int sy  = y0 - 2 + r;
                int sx  = x0 - 4 + s4 * 4;
                // segments 4-aligned; 112 % 4 == 0 -> fully in or fully out
                if ((unsigned)sy < (unsigned)HW && (unsigned)sx <= (unsigned)(HW - 4)) {
                    const float* gp = src + (size_t)d * LPIX + sy * HW + sx;
                    lds_fp lp = (lds_fp)&dst[(d * CH + r) * CW + s4 * 4];
                    asm volatile("global_load_async_to_lds_b128 %0, %1, off"
                                 :: "v"(lp), "v"(gp) : "memory");
                }
            }
        }
        asm volatile("s_wait_asynccnt 0x0" ::: "memory");
        __syncthreads();
    }

    // ---- per-thread attention for one pixel ----
    // logits over the 9 dilated neighbors
    float lg[9];
    #pragma unroll
    for (int jy = 0; jy < 3; ++jy) {
        #pragma unroll
        for (int jx = 0; jx < 3; ++jx) {
            // neighbor (gy + 2jy-2, gx + 2jx-2) -> container (ty+2jy, tx+2jx+2)
            const float* kb = &smem[(ty + 2 * jy) * CW + (tx + 2 * jx + 2)];
            float acc = 0.f;
            #pragma unroll
            for (int d = 0; d < HD; ++d)
                acc = __builtin_fmaf(qr[d], kb[d * (CH * CW)], acc);
            lg[jy * 3 + jx] = acc * SCALE;
        }
    }

    // softmax over 9 (zero-padded OOB logits participate, matching reference)
    float m = lg[0];
    #pragma unroll
    for (int j = 1; j < 9; ++j) m = fmaxf(m, lg[j]);
    float s = 0.f;
    #pragma unroll
    for (int j = 0; j < 9; ++j) { lg[j] = __expf(lg[j] - m); s += lg[j]; }
    const float inv = 1.0f / s;

    // weighted sum over window from V tile
    float acc[HD];
    #pragma unroll
    for (int d = 0; d < HD; ++d) acc[d] = 0.f;
    #pragma unroll
    for (int jy = 0; jy < 3; ++jy) {
        #pragma unroll
        for (int jx = 0; jx < 3; ++jx) {
            const float w = lg[jy * 3 + jx] * inv;
            const float* vb = &smem[TILE_F + (ty + 2 * jy) * CW + (tx + 2 * jx + 2)];
            #pragma unroll
            for (int d = 0; d < HD; ++d)
                acc[d] = __builtin_fmaf(w, vb[d * (CH * CW)], acc[d]);
        }
    }

    // output layout [B, H, W, heads*HD]; 128 contiguous bytes per thread
    float* op = out + ((size_t)((b * HW + gy) * HW + gx)) * (HEADS * HD) + head * HD;
    #pragma unroll
    for (int d = 0; d < HD; d += 4)
        *(float4*)&op[d] = make_float4(acc[d], acc[d + 1], acc[d + 2], acc[d + 3]);
}

extern "C" void kernel_launch(void* const* d_in, const int* in_sizes, int n_in,
                              void* d_out, int out_size, void* d_ws, size_t ws_size,
                              hipStream_t stream)
{
    const float* q = (const float*)d_in[0];
    const float* k = (const float*)d_in[1];
    const float* v = (const float*)d_in[2];
    float* out = (float*)d_out;

    const int B = in_sizes[0] / (HEADS * HD * HW * HW);   // = 4
    dim3 grid(HW / TS, HW / TS, B * HEADS);               // 7 x 7 x 24 = 1176 blocks
    dim3 block(256);
    dilate_attn_kernel<<<grid, block, 0, stream>>>(q, k, v, out);
}